// HierarchicalSampler_30528627539971
// MI455X (gfx1250) — compile-verified
//
#include <hip/hip_runtime.h>

typedef __attribute__((ext_vector_type(2))) float v2f;
typedef __attribute__((ext_vector_type(8))) float v8f;

#define NC 64
#define NS 64
#define NT (NC + NS)   // 128 merged samples per ray

// Branchless rank over a 64-entry ascending LDS array.
// strict=true  -> count of a[i] <  x
// strict=false -> count of a[i] <= x   (== searchsorted side='right')
__device__ __forceinline__ int rank64(const float* a, float x, bool strict) {
  int pos = 0;
#pragma unroll
  for (int s = 32; s >= 1; s >>= 1) {
    float v = a[pos + s - 1];
    bool p = strict ? (v < x) : (v <= x);
    pos += p ? s : 0;
  }
  float v = a[pos];               // pos <= 63 here
  bool p = strict ? (v < x) : (v <= x);
  return pos + (p ? 1 : 0);       // result in [0, 64]
}

__global__ __launch_bounds__(32) void nerf_hier_sampler(
    const float* __restrict__ origins,
    const float* __restrict__ directions,
    const float* __restrict__ z_coarse,
    const float* __restrict__ weights,
    const float* __restrict__ uin,
    float* __restrict__ out,
    int B)
{
  __shared__ float s_zc[NC];
  __shared__ float s_wt[NC];
  __shared__ float s_cdf[NC];
  __shared__ float s_zf[NS];
  __shared__ float s_zm[NT];
  __shared__ __align__(16) float s_pts[NT * 3];
  __shared__ float s_scan[32];

  const int ray = blockIdx.x;
  if (ray >= B) return;
  const int lane = threadIdx.x;
  const size_t rb = (size_t)ray * NC;

  // ---- stage z_coarse / weights into LDS (coalesced b64) ----
  float2 z2 = ((const float2*)(z_coarse + rb))[lane];
  ((float2*)s_zc)[lane] = z2;
  float2 w2 = ((const float2*)(weights + rb))[lane];
  ((float2*)s_wt)[lane] = w2;
  __syncthreads();

  // ---- midpoint weights (+eps), pairwise scan -> CDF ----
  const int i0 = 2 * lane;
  float p0 = 0.5f * (s_wt[i0] + s_wt[i0 + 1]) + 1e-5f;                 // i in [0,62]
  float p1 = (i0 + 1 < 63) ? (0.5f * (s_wt[i0 + 1] + s_wt[i0 + 2]) + 1e-5f)
                           : 0.0f;                                      // pad index 63
  float pair = p0 + p1;
  s_scan[lane] = pair;
  __syncthreads();
#pragma unroll
  for (int d = 1; d < 32; d <<= 1) {
    float t = (lane >= d) ? s_scan[lane - d] : 0.0f;
    __syncthreads();
    s_scan[lane] += t;
    __syncthreads();
  }
  float incl  = s_scan[lane];
  float total = s_scan[31];
  float excl  = incl - pair;
  float inv   = 1.0f / (total + 1e-8f);                 // pdf = w/(sum+1e-8); cumsum commutes
  float2 c2;  c2.x = excl * inv;  c2.y = (excl + p0) * inv;
  ((float2*)s_cdf)[lane] = c2;                          // cdf[0]=0 ... cdf[63]=~1
  __syncthreads();

  // ---- invert CDF: 2 fine samples per lane ----
  const float* uR = uin + (size_t)ray * NS;
#pragma unroll
  for (int h = 0; h < 2; ++h) {
    float uu = uR[lane + 32 * h];
    uu = fminf(fmaxf(uu, 1e-5f), 1.0f - 1e-5f);
    int idx   = rank64(s_cdf, uu, false);               // searchsorted 'right'
    int below = idx - 1; below = below < 0 ? 0 : below; // idx>=1 since cdf[0]=0<u
    int above = idx > 63 ? 63 : idx;
    float cl = s_cdf[below], ch = s_cdf[above];
    float bl = s_zc[below],  bh = s_zc[above];
    float den = ch - cl;
    den = (den < 1e-5f) ? 1.0f : den;
    float t = (uu - cl) / den;
    s_zf[lane + 32 * h] = bl + t * (bh - bl);
  }
  __syncthreads();

  // ---- bitonic sort of the 64 fine depths (32 lanes x 1 pair/round) ----
#pragma unroll
  for (int k = 2; k <= 64; k <<= 1) {
#pragma unroll
    for (int j = k >> 1; j > 0; j >>= 1) {
      int low = lane & (j - 1);
      int i   = (lane << 1) - low;                      // index with bit j clear
      float a = s_zf[i];
      float b = s_zf[i + j];
      bool asc = (i & k) == 0;
      float mn = fminf(a, b), mx = fmaxf(a, b);
      s_zf[i]     = asc ? mn : mx;
      s_zf[i + j] = asc ? mx : mn;
      __syncthreads();
    }
  }

  // ---- merge sorted coarse + sorted fine via rank scatter ----
#pragma unroll
  for (int h = 0; h < 2; ++h) {
    int i = lane + 32 * h;
    float vc = s_zc[i];
    s_zm[i + rank64(s_zf, vc, true)]  = vc;             // coarse before equal fine
    float vf = s_zf[i];
    s_zm[i + rank64(s_zc, vf, false)] = vf;             // fine after equal coarse
  }
  __syncthreads();

  // ---- ray origin/direction (block-uniform -> scalar loads) ----
  const float* o  = origins    + (size_t)ray * 3;
  const float* dv = directions + (size_t)ray * 3;
  float o0 = o[0],  o1 = o[1],  o2 = o[2];
  float d0 = dv[0], d1 = dv[1], d2 = dv[2];

  // ---- point expansion via V_WMMA_F32_16X16X4_F32:
  //      D[M][N] = z_M * d_N + 1 * o_N   with A = [z | 1 | 0 | 0], B = [d; o; 0; 0]
  // B layout (4x16 f32): VGPR0 = K0 (lanes 0-15) / K2 (lanes 16-31); VGPR1 = K1 / K3.
  float bk0 = (lane == 0) ? d0 : (lane == 1) ? d1 : (lane == 2) ? d2 : 0.0f;
  float bk1 = (lane == 0) ? o0 : (lane == 1) ? o1 : (lane == 2) ? o2 : 0.0f;
  v2f Bv;  Bv[0] = bk0;  Bv[1] = bk1;                   // lanes>=16 give K2=K3=0

  const int col = lane & 15;                            // D: lane -> column N
  const int mb  = (lane >> 4) << 3;                     // D: VGPR v -> row M = mb+v
#pragma unroll
  for (int t = 0; t < 8; ++t) {
    // A layout (16x4 f32): VGPR0 = K0 (lanes 0-15, M=lane) / K2 (lanes 16-31);
    //                      VGPR1 = K1 / K3.  Zero the K2/K3 halves.
    float za = (lane < 16) ? s_zm[t * 16 + lane] : 0.0f;
    float a1 = (lane < 16) ? 1.0f : 0.0f;
    v2f Av;  Av[0] = za;  Av[1] = a1;
    v8f Cv = {};
    v8f D = __builtin_amdgcn_wmma_f32_16x16x4_f32(false, Av, false, Bv,
                                                  (short)0, Cv, false, false);
    if (col < 3) {
#pragma unroll
      for (int vv = 0; vv < 8; ++vv)
        s_pts[(t * 16 + mb + vv) * 3 + col] = D[vv];
    }
  }
  __syncthreads();

  // ---- fully coalesced b128 output stores: 3 x float4 per lane ----
  float4* outv = (float4*)(out + (size_t)ray * (NT * 3));
#pragma unroll
  for (int r = 0; r < 3; ++r) {
    int f4 = lane + 32 * r;
    const float* p = s_pts + 4 * f4;
    float4 val;  val.x = p[0];  val.y = p[1];  val.z = p[2];  val.w = p[3];
    outv[f4] = val;
  }
}

extern "C" void kernel_launch(void* const* d_in, const int* in_sizes, int n_in,
                              void* d_out, int out_size, void* d_ws, size_t ws_size,
                              hipStream_t stream) {
  const float* origins    = (const float*)d_in[0];
  const float* directions = (const float*)d_in[1];
  const float* z_coarse   = (const float*)d_in[2];
  const float* weights    = (const float*)d_in[3];
  const float* uin        = (const float*)d_in[4];
  float* out = (float*)d_out;
  const int B = in_sizes[0] / 3;

  hipLaunchKernelGGL(nerf_hier_sampler, dim3(B), dim3(32), 0, stream,
                     origins, directions, z_coarse, weights, uin, out, B);
}